// GCN_CORA_36644660969781
// MI455X (gfx1250) — compile-verified
//
#include <hip/hip_runtime.h>

typedef float v2f __attribute__((ext_vector_type(2)));
typedef float v4f __attribute__((ext_vector_type(4)));
typedef float v8f __attribute__((ext_vector_type(8)));
typedef int   async_v4i __attribute__((vector_size(16)));   // matches builtin param type

// ---------------------------------------------------------------------------
// Async global->LDS copy (CDNA5 GLOBAL_LOAD_ASYNC_TO_LDS_B128, ASYNCcnt).
// Builtin signature (from hipcc diagnostic): param0 = AS1 int4*, so we pass
// AS1/AS3 int4 pointers.
// ---------------------------------------------------------------------------
#if defined(__has_builtin)
#  if __has_builtin(__builtin_amdgcn_global_load_async_to_lds_b128) && \
      __has_builtin(__builtin_amdgcn_s_wait_asynccnt)
#    define HAVE_ASYNC_LDS 1
#  endif
#endif
#ifndef HAVE_ASYNC_LDS
#  define HAVE_ASYNC_LDS 0
#endif

__device__ inline void cp16_to_lds(const float* __restrict__ g, float* l) {
#if HAVE_ASYNC_LDS
    __builtin_amdgcn_global_load_async_to_lds_b128(
        (__attribute__((address_space(1))) async_v4i*)g,
        (__attribute__((address_space(3))) async_v4i*)l,
        /*imm offset=*/0, /*cpol=*/0);
#else
    *(v4f*)l = *(const v4f*)g;
#endif
}

__device__ inline void wait_async_copies() {
#if HAVE_ASYNC_LDS
    __builtin_amdgcn_s_wait_asynccnt(0);
#endif
}

// ---------------------------------------------------------------------------
// WMMA helper: D = A(16x4 f32) * B(4x16 f32) + C(16x16 f32), wave32.
// ---------------------------------------------------------------------------
__device__ inline v8f wmma_f32_k4(v2f a, v2f b, v8f c) {
    return __builtin_amdgcn_wmma_f32_16x16x4_f32(
        false, a, false, b, (short)0, c, false, false);
}

// ---------------------------------------------------------------------------
// Degree / normalization kernels
// ---------------------------------------------------------------------------
__global__ void k_deg_init(float* __restrict__ deg, int N) {
    int i = blockIdx.x * blockDim.x + threadIdx.x;
    if (i < N) deg[i] = 1.0f;  // self-loop
}

__global__ void k_deg_count(float* __restrict__ deg, const int* __restrict__ dst, int E) {
    int e = blockIdx.x * blockDim.x + threadIdx.x;
    if (e < E) atomicAdd(&deg[dst[e]], 1.0f);
}

__global__ void k_dinv(float* __restrict__ deg, int N) {
    int i = blockIdx.x * blockDim.x + threadIdx.x;
    if (i < N) deg[i] = rsqrtf(deg[i]);   // deg >= 1 always (self-loops)
}

// ---------------------------------------------------------------------------
// GEMM1: H1[N,16] = X[N,K] @ W1[K,16] via V_WMMA_F32_16X16X4_F32.
// 128 threads = 4 waves; each wave owns a 16-row C tile (64 rows / block).
// X and W1 are staged through double-buffered LDS with async b128 copies;
// the K-tail is zero-padded during staging so the compute loop is guard-free.
// A LDS pitch 98 dwords -> conflict-free ds_load_b64 A-fragment reads.
// Fragment layouts per ISA tables:
//   A 16x4: lanes 0-15 rows, v0/v1 = K+0/K+1; lanes 16-31 rows, K+2/K+3.
//   B 4x16: v0 = {K+0 | K+2}, v1 = {K+1 | K+3}; col = lane & 15.
//   C 16x16: VGPR v -> row v (lanes 0-15) / v+8 (lanes 16-31); col = lane & 15.
// ---------------------------------------------------------------------------
#define KCHUNK 96
#define APITCH 98
#define GROWS  64
#define GT     128

__global__ __launch_bounds__(GT) void k_gemm1(const float* __restrict__ X,
                                              const float* __restrict__ W1,
                                              float* __restrict__ H1,
                                              int N, int K) {
    __shared__ float sA[2][GROWS * APITCH];   // 50176 B
    __shared__ float sB[2][KCHUNK * 16];      // 12288 B

    const int lane  = threadIdx.x & 31;
    const int wave  = threadIdx.x >> 5;
    const int m     = lane & 15;
    const int khalf = (lane < 16) ? 0 : 2;
    const int rblk  = blockIdx.x * GROWS;
    const int rloc  = wave * 16 + m;

    const int nchunks = (K + KCHUNK - 1) / KCHUNK;

    auto stage = [&](int buf, int k0) {
        if (k0 + KCHUNK <= K) {
            // Full chunk: guard-free async copies, fully coalesced.
            for (int j = threadIdx.x; j < KCHUNK * 4; j += GT)        // B panel
                cp16_to_lds(W1 + (size_t)k0 * 16 + (size_t)j * 4, &sB[buf][j * 4]);
            for (int j = threadIdx.x; j < GROWS * (KCHUNK / 4); j += GT) {  // A tile
                int r  = j / (KCHUNK / 4);
                int kq = j - r * (KCHUNK / 4);
                int gr = rblk + r; if (gr >= N) gr = N - 1;           // clamp, masked at store
                cp16_to_lds(X + (size_t)gr * K + k0 + kq * 4,
                            &sA[buf][r * APITCH + kq * 4]);
            }
        } else {
            // Tail chunk (once): clamped loads + zero pad, no exec games needed after.
            for (int i = threadIdx.x; i < KCHUNK * 16; i += GT) {
                int k = k0 + (i >> 4);
                float v = W1[(size_t)((k < K) ? k : K - 1) * 16 + (i & 15)];
                sB[buf][i] = (k < K) ? v : 0.0f;
            }
            for (int i = threadIdx.x; i < GROWS * KCHUNK; i += GT) {
                int r  = i / KCHUNK;
                int kk = i - r * KCHUNK;
                int gr = rblk + r; if (gr >= N) gr = N - 1;
                int k  = k0 + kk;
                float v = X[(size_t)gr * K + ((k < K) ? k : K - 1)];
                sA[buf][r * APITCH + kk] = (k < K) ? v : 0.0f;
            }
        }
    };

    stage(0, 0);
    v8f acc = {};
    for (int c = 0; c < nchunks; ++c) {
        wait_async_copies();       // this wave's async stores have landed
        __syncthreads();           // everyone's have; buffer c&1 is ready
        if (c + 1 < nchunks) stage((c + 1) & 1, (c + 1) * KCHUNK);  // overlap DMA
        const float* __restrict__ A = sA[c & 1];
        const float* __restrict__ B = sB[c & 1];
        #pragma unroll
        for (int kk = 0; kk < KCHUNK; kk += 4) {
            v2f a = *(const v2f*)&A[rloc * APITCH + kk + khalf];
            v2f b;
            b.x = B[(kk + khalf)     * 16 + m];
            b.y = B[(kk + khalf + 1) * 16 + m];
            acc = wmma_f32_k4(a, b, acc);
        }
    }

    const int crow0 = rblk + wave * 16 + ((lane < 16) ? 0 : 8);
    #pragma unroll
    for (int v = 0; v < 8; ++v) {
        int r = crow0 + v;
        if (r < N) H1[(size_t)r * 16 + m] = acc[v];
    }
}

// ---------------------------------------------------------------------------
// GEMM2: H2[N,8pad] = A1[N,16] @ W2pad[16,16] (cols >= FOUT zero). K=16.
// ---------------------------------------------------------------------------
__global__ __launch_bounds__(256) void k_gemm2(const float* __restrict__ A1,
                                               const float* __restrict__ W2,
                                               float* __restrict__ H2,
                                               int N, int FOUT) {
    __shared__ float sB[16 * 16];
    {
        int i = threadIdx.x;              // exactly 256 entries
        int k = i >> 4, n = i & 15;
        sB[i] = (n < FOUT) ? W2[k * FOUT + n] : 0.0f;
    }
    __syncthreads();

    const int lane  = threadIdx.x & 31;
    const int wave  = threadIdx.x >> 5;
    const int rbase = (blockIdx.x * 8 + wave) * 16;
    const int m     = lane & 15;
    const int khalf = (lane < 16) ? 0 : 2;

    int arow = rbase + m;
    if (arow >= N) arow = N - 1;
    const float* __restrict__ ar = A1 + (size_t)arow * 16;

    v8f c = {};
    #pragma unroll
    for (int kk = 0; kk < 16; kk += 4) {
        v2f a = *(const v2f*)&ar[kk + khalf];
        v2f b;
        b.x = sB[(kk + khalf)     * 16 + m];
        b.y = sB[(kk + khalf + 1) * 16 + m];
        c = wmma_f32_k4(a, b, c);
    }

    const int crow0 = rbase + ((lane < 16) ? 0 : 8);
    #pragma unroll
    for (int v = 0; v < 8; ++v) {
        int r = crow0 + v;
        if (r < N && m < 8) H2[(size_t)r * 8 + m] = c[v];   // col 7 is zero pad
    }
}

// ---------------------------------------------------------------------------
// Aggregation: self-loop seed + atomic edge scatter. Width = 1<<s features.
// ---------------------------------------------------------------------------
__global__ void k_self(const float* __restrict__ h, float* __restrict__ a,
                       const float* __restrict__ dinv, int N, int s) {
    int tid = blockIdx.x * blockDim.x + threadIdx.x;
    if (tid >= (N << s)) return;
    int n = tid >> s;
    float d = dinv[n];
    a[tid] = d * d * h[tid];
}

__global__ void k_edge_scatter(const float* __restrict__ h, float* __restrict__ a,
                               const float* __restrict__ dinv,
                               const int* __restrict__ src, const int* __restrict__ dst,
                               int E, int s) {
    int tid = blockIdx.x * blockDim.x + threadIdx.x;
    if (tid >= (E << s)) return;
    int e = tid >> s;
    int f = tid & ((1 << s) - 1);
    int u = src[e], v = dst[e];
    float norm = dinv[u] * dinv[v];
    atomicAdd(&a[((size_t)v << s) + f], h[((size_t)u << s) + f] * norm);
}

__global__ void k_bias_relu(float* __restrict__ a, const float* __restrict__ b, int N) {
    int tid = blockIdx.x * blockDim.x + threadIdx.x;
    if (tid >= N * 16) return;
    float v = a[tid] + b[tid & 15];
    a[tid] = v > 0.0f ? v : 0.0f;
}

__global__ void k_logsoftmax(const float* __restrict__ a2, const float* __restrict__ b2,
                             float* __restrict__ out, int N, int FOUT) {
    int n = blockIdx.x * blockDim.x + threadIdx.x;
    if (n >= N) return;
    float z[8];
    float mx = -3.402823466e+38f;
    for (int c = 0; c < FOUT; ++c) {
        z[c] = a2[(size_t)n * 8 + c] + b2[c];
        mx = fmaxf(mx, z[c]);
    }
    float se = 0.0f;
    for (int c = 0; c < FOUT; ++c) se += __expf(z[c] - mx);
    float lse = __logf(se);
    for (int c = 0; c < FOUT; ++c) out[(size_t)n * FOUT + c] = z[c] - mx - lse;
}

// ---------------------------------------------------------------------------
// Host launcher
// ---------------------------------------------------------------------------
extern "C" void kernel_launch(void* const* d_in, const int* in_sizes, int n_in,
                              void* d_out, int out_size, void* d_ws, size_t ws_size,
                              hipStream_t stream) {
    const float* x  = (const float*)d_in[0];
    const int*   ei = (const int*)d_in[1];
    const float* W1 = (const float*)d_in[2];
    const float* b1 = (const float*)d_in[3];
    const float* W2 = (const float*)d_in[4];
    const float* b2 = (const float*)d_in[5];
    float* out = (float*)d_out;

    const int FHID = in_sizes[3];           // 16
    const int FOUT = in_sizes[5];           // 7
    const int FIN  = in_sizes[2] / FHID;    // 1433
    const int N    = in_sizes[0] / FIN;     // 100000
    const int E    = in_sizes[1] / 2;       // 3200000
    const int* src = ei;
    const int* dst = ei + E;

    float* ws  = (float*)d_ws;
    float* deg = ws;                        // N floats (becomes dinv in place)
    float* h1  = deg + N;                   // N*16
    float* a1  = h1 + (size_t)N * 16;       // N*16
    float* h2  = a1 + (size_t)N * 16;       // N*8 (padded)
    float* a2  = h2 + (size_t)N * 8;        // N*8 (padded)

    const int T = 256;
    auto cdiv = [](long long a, long long b) { return (int)((a + b - 1) / b); };

    // degrees + symmetric normalization
    k_deg_init  <<<cdiv(N, T), T, 0, stream>>>(deg, N);
    k_deg_count <<<cdiv(E, T), T, 0, stream>>>(deg, dst, E);
    k_dinv      <<<cdiv(N, T), T, 0, stream>>>(deg, N);

    // layer 1
    k_gemm1        <<<cdiv(N, GROWS), GT, 0, stream>>>(x, W1, h1, N, FIN);
    k_self         <<<cdiv((long long)N * 16, T), T, 0, stream>>>(h1, a1, deg, N, 4);
    k_edge_scatter <<<cdiv((long long)E * 16, T), T, 0, stream>>>(h1, a1, deg, src, dst, E, 4);
    k_bias_relu    <<<cdiv((long long)N * 16, T), T, 0, stream>>>(a1, b1, N);

    // layer 2
    k_gemm2        <<<cdiv(N, 128), T, 0, stream>>>(a1, W2, h2, N, FOUT);
    k_self         <<<cdiv((long long)N * 8, T),  T, 0, stream>>>(h2, a2, deg, N, 3);
    k_edge_scatter <<<cdiv((long long)E * 8, T),  T, 0, stream>>>(h2, a2, deg, src, dst, E, 3);

    // output
    k_logsoftmax   <<<cdiv(N, T), T, 0, stream>>>(a2, b2, out, N, FOUT);
}